// LTC_16372415332469
// MI455X (gfx1250) — compile-verified
//
#include <hip/hip_runtime.h>
#include <hip/hip_bf16.h>

// LTC forward for MI455X (gfx1250). B=32,T=64,IN=64,U=128,MOTOR=32,UNFOLDS=6.
// Design: latency-bound sequential scan -> 1 workgroup per batch row (32 WGPs),
// 512 threads = 4-way j-split x 128 post-units. Recurrent weights packed as
// float4 {sigma*log2e, sigma*mu*log2e, w*mask, w*mask*erev} and staged into
// 256KB of LDS via the Tensor Data Mover (CDNA5 TDM). Output affine map
// (a diagonal matmul) done with v_wmma_f32_16x16x4_f32 tiles.

static constexpr int kB = 32, kT = 64, kIN = 64, kU = 128, kMOTOR = 32, kUNF = 6;
static constexpr float kEPS = 1e-8f;
static constexpr float kLOG2E = 1.44269504088896340736f;

typedef float  v2f  __attribute__((ext_vector_type(2)));
typedef float  v8f  __attribute__((ext_vector_type(8)));
typedef unsigned int v4ui __attribute__((ext_vector_type(4)));
typedef int    v8si __attribute__((ext_vector_type(8)));
typedef int    v4si __attribute__((ext_vector_type(4)));

// ---------------------------------------------------------------- pack kernel
// rec4[j*U+u] = {sigma*log2e, sigma*mu*log2e, w*mask, w*mask*erev}
// sens4[i*U+u] likewise for the sensory synapses.
__global__ void ltc_pack(const float* __restrict__ sigma, const float* __restrict__ mu,
                         const float* __restrict__ w,     const float* __restrict__ erev,
                         const float* __restrict__ mask,
                         const float* __restrict__ ssig,  const float* __restrict__ smu,
                         const float* __restrict__ sw,    const float* __restrict__ serev,
                         const float* __restrict__ smask,
                         float4* __restrict__ rec4, float4* __restrict__ sens4) {
  int i = blockIdx.x * blockDim.x + threadIdx.x;
  if (i < kU * kU) {
    float sc = sigma[i] * kLOG2E;
    float wm = w[i] * mask[i];
    rec4[i] = make_float4(sc, sc * mu[i], wm, wm * erev[i]);
  }
  if (i < kIN * kU) {
    float sc = ssig[i] * kLOG2E;
    float wm = sw[i] * smask[i];
    sens4[i] = make_float4(sc, sc * smu[i], wm, wm * serev[i]);
  }
}

// ---------------------------------------------------------------- scan kernel
// grid = B blocks, block = 512 threads: tid = r*128 + u, r in [0,4) owns j/i slice.
__global__ __launch_bounds__(512)
void ltc_scan(const float4* __restrict__ gRec4, const float4* __restrict__ gSens4,
              const float* __restrict__ inseq, const float* __restrict__ tspan,
              const float* __restrict__ input_w, const float* __restrict__ input_b,
              const float* __restrict__ gleak, const float* __restrict__ vleak,
              const float* __restrict__ cm, float* __restrict__ out) {
  extern __shared__ float smem[];
  float4* sRec  = (float4*)smem;              // 16384 * float4 = 256 KB
  float*  sV    = smem + 4 * 16384;           // 128 floats
  float2* sPart = (float2*)(sV + kU);         // 512 * float2 = 4 KB

  const int b   = blockIdx.x;
  const int tid = threadIdx.x;
  const int u   = tid & (kU - 1);
  const int r   = tid >> 7;

  // --- TDM: DMA the 256KB packed recurrent weights global -> LDS (wave 0). ---
  // D# describes a 2-D f32 tile 8192 x 8 (== 65536 elements), stride0 = 8192.
  if (tid < 32) {
    unsigned lds0 = __builtin_amdgcn_groupstaticsize();   // dynamic LDS base
    unsigned long long ga = (unsigned long long)(const void*)gRec4;
    v4ui g0 = { 1u,                                   // count=1, user mode
                lds0,                                 // lds_addr
                (unsigned)ga,                         // global_addr[31:0]
                (unsigned)((ga >> 32) & 0x1FFFFFFull) // global_addr[56:32]
                  | (2u << 30) };                     // type=2 ("image")
    v8si g1 = { (int)(2u << 16),          // data_size=2 (4B), mask=0, flags=0
                (int)(0x2000u << 16),     // tensor_dim0[15:0]=8192
                (int)(8u << 16),          // tensor_dim0 hi=0, tensor_dim1=8
                (int)(0x2000u << 16),     // tensor_dim1 hi=0, tile_dim0=8192
                8,                        // tile_dim1=8, tile_dim2=0
                0x2000,                   // tensor_dim0_stride lo = 8192
                0, 0 };                   // stride0 hi / stride1 = 0
    v4si gz4 = { 0, 0, 0, 0 };            // groups 2/3 unused (<=2D tensor)
    v8si gz8 = { 0, 0, 0, 0, 0, 0, 0, 0 };
    __builtin_amdgcn_tensor_load_to_lds(g0, g1, gz4, gz4, gz8, 0);
    __builtin_amdgcn_s_wait_tensorcnt(0);
  }

  float cm_u = 0.f, gl_u = 0.f, glvl = 0.f;
  if (r == 0) {
    cm_u = cm[u];
    gl_u = gleak[u];
    glvl = gl_u * vleak[u];
    sV[u] = 0.f;
  }
  float v_reg = 0.f;
  __syncthreads();   // TDM data + sV init visible to all waves

  for (int t = 0; t < kT; ++t) {
    // ---- sensory synapses: partial sums over this thread's 16 inputs ----
    float sn = 0.f, sd = 0.f;
    const float* xrow = inseq + (b * kT + t) * kIN;
    const int i0 = r * 16;
    #pragma unroll
    for (int ii = 0; ii < 16; ++ii) {
      const int i = i0 + ii;
      float x = fmaf(xrow[i], input_w[i], input_b[i]);
      float4 f = gSens4[i * kU + u];
      // sigmoid(sigma*(x-mu)) = 1 / (1 + 2^(sc*mu - sc*x))
      float e = __builtin_amdgcn_exp2f(fmaf(-f.x, x, f.y));
      float s = __builtin_amdgcn_rcpf(1.0f + e);
      sn = fmaf(f.w, s, sn);   // * w*mask*erev
      sd = fmaf(f.z, s, sd);   // * w*mask
    }
    sPart[r * kU + u] = make_float2(sn, sd);
    __syncthreads();

    float wnum_s = 0.f, wden_s = 0.f, cmt = 0.f;
    if (r == 0) {
      float2 p0 = sPart[u], p1 = sPart[kU + u], p2 = sPart[2 * kU + u], p3 = sPart[3 * kU + u];
      wnum_s = p0.x + p1.x + p2.x + p3.x;
      wden_s = p0.y + p1.y + p2.y + p3.y;
      cmt = cm_u * (float)kUNF / tspan[b * kT + t];   // cm / (t/unfolds)
    }
    __syncthreads();   // sPart reusable

    // ---- 6 ODE unfolds ----
    for (int k = 0; k < kUNF; ++k) {
      float pn = 0.f, pd = 0.f;
      const int j0 = r * 32;
      #pragma unroll 8
      for (int jj = 0; jj < 32; ++jj) {
        const int j = j0 + jj;
        float vj = sV[j];
        float4 f = sRec[j * kU + u];           // ds_load_b128 from LDS
        float e = __builtin_amdgcn_exp2f(fmaf(-f.x, vj, f.y));
        float s = __builtin_amdgcn_rcpf(1.0f + e);
        pn = fmaf(f.w, s, pn);
        pd = fmaf(f.z, s, pd);
      }
      sPart[r * kU + u] = make_float2(pn, pd);
      __syncthreads();
      if (r == 0) {
        float2 p0 = sPart[u], p1 = sPart[kU + u], p2 = sPart[2 * kU + u], p3 = sPart[3 * kU + u];
        float wnum = p0.x + p1.x + p2.x + p3.x + wnum_s;
        float wden = p0.y + p1.y + p2.y + p3.y + wden_s;
        v_reg = (cmt * v_reg + glvl + wnum) / (cmt + gl_u + wden + kEPS);
        sV[u] = v_reg;
      }
      __syncthreads();
    }

    // raw v for motor neurons; output affine applied by the WMMA pass
    if (r == 0 && u < kMOTOR) out[(b * kT + t) * kMOTOR + u] = v_reg;
  }
  if (r == 0) out[kB * kT * kMOTOR + b * kU + u] = v_reg;   // vT
}

// ------------------------------------------------------- output affine (WMMA)
// out[bt, n] = V[bt, n]*output_w[n] + output_b[n]  ==  V x diag(output_w) + B.
// Per wave: one 16x16 tile of out, K=16 split into 4 chained v_wmma_f32_16x16x4_f32.
// Only diagonal 16x16 blocks of the 32x32 diag matrix are nonzero -> 2 n-blocks.
__global__ __launch_bounds__(256)
void ltc_output_affine(float* __restrict__ out,
                       const float* __restrict__ ow, const float* __restrict__ ob) {
  const int wave = threadIdx.x >> 5;
  const int lane = threadIdx.x & 31;
  const int tile = blockIdx.x * (blockDim.x >> 5) + wave;  // 0..255
  const int bt0  = (tile >> 1) * 16;                       // 128 bt-tiles
  const int nblk = tile & 1;                               // 2 motor halves
  const int half = lane >> 4;                              // hi/lo lane group
  const int l    = lane & 15;
  const int n    = nblk * 16 + l;

  const float w_n = ow[n];
  const float b_n = ob[n];

  v8f acc;
  #pragma unroll
  for (int q = 0; q < 8; ++q) acc[q] = b_n;       // C = broadcast bias

  const float* Vrow = out + (bt0 + l) * kMOTOR + nblk * 16;
  #pragma unroll
  for (int c = 0; c < 4; ++c) {
    // A 16x4 f32: lanes 0-15 rows M=lane {K=c*4+0, c*4+1}; lanes 16-31 {K=+2,+3}
    const int ka = c * 4 + (half ? 2 : 0);
    const int kb = c * 4 + (half ? 3 : 1);
    v2f A; A.x = Vrow[ka]; A.y = Vrow[kb];
    // B 4x16 diag fragment: B[kk, l] = (global k == l) ? output_w : 0
    v2f Bf;
    Bf.x = (ka == l) ? w_n : 0.0f;
    Bf.y = (kb == l) ? w_n : 0.0f;
    acc = __builtin_amdgcn_wmma_f32_16x16x4_f32(
        /*neg_a=*/false, A, /*neg_b=*/false, Bf,
        /*c_mod=*/(short)0, acc, /*reuse_a=*/false, /*reuse_b=*/false);
  }
  // D 16x16: VGPR q -> rows M=q (lanes 0-15) and M=q+8 (lanes 16-31), N=l
  #pragma unroll
  for (int q = 0; q < 8; ++q) {
    const int m = q + half * 8;
    out[(bt0 + m) * kMOTOR + n] = acc[q];
  }
}

// --------------------------------------------------------------------- launch
extern "C" void kernel_launch(void* const* d_in, const int* in_sizes, int n_in,
                              void* d_out, int out_size, void* d_ws, size_t ws_size,
                              hipStream_t stream) {
  (void)in_sizes; (void)n_in; (void)out_size; (void)ws_size;
  const float* inseq   = (const float*)d_in[0];
  const float* tspan   = (const float*)d_in[1];
  const float* iw      = (const float*)d_in[2];
  const float* ib      = (const float*)d_in[3];
  const float* gleak   = (const float*)d_in[4];
  const float* vleak   = (const float*)d_in[5];
  const float* cmv     = (const float*)d_in[6];
  const float* sigma   = (const float*)d_in[7];
  const float* mu      = (const float*)d_in[8];
  const float* w       = (const float*)d_in[9];
  const float* erev    = (const float*)d_in[10];
  const float* mask    = (const float*)d_in[11];
  const float* ssig    = (const float*)d_in[12];
  const float* smu     = (const float*)d_in[13];
  const float* sw      = (const float*)d_in[14];
  const float* serev   = (const float*)d_in[15];
  const float* smask   = (const float*)d_in[16];
  const float* outw    = (const float*)d_in[17];
  const float* outb    = (const float*)d_in[18];
  float* out = (float*)d_out;

  float4* rec4  = (float4*)d_ws;                         // 256 KB
  float4* sens4 = (float4*)((float*)d_ws + 4 * 16384);   // 128 KB

  ltc_pack<<<(kU * kU + 255) / 256, 256, 0, stream>>>(
      sigma, mu, w, erev, mask, ssig, smu, sw, serev, smask, rec4, sens4);

  const size_t ldsBytes = (size_t)(4 * 16384 + kU) * sizeof(float)
                        + 4 * kU * sizeof(float2);       // 266,752 B < 320 KB
  (void)hipFuncSetAttribute(reinterpret_cast<const void*>(ltc_scan),
                            hipFuncAttributeMaxDynamicSharedMemorySize, (int)ldsBytes);

  ltc_scan<<<kB, 512, ldsBytes, stream>>>(rec4, sens4, inseq, tspan, iw, ib,
                                          gleak, vleak, cmv, out);

  ltc_output_affine<<<32, 256, 0, stream>>>(out, outw, outb);
}